// EPS_68771016344265
// MI455X (gfx1250) — compile-verified
//
#include <hip/hip_runtime.h>

typedef __attribute__((ext_vector_type(2))) float v2f;
typedef __attribute__((ext_vector_type(4))) float v4f;
typedef __attribute__((ext_vector_type(8))) float v8f;

namespace {
constexpr int kB     = 128;
constexpr int kH     = 128;
constexpr int kW     = 128;
constexpr int kS     = 2;
constexpr int kOut   = 4;
constexpr int kHo    = kH - 1;   // 127 output rows
constexpr int kWo    = kW - 1;   // 127 output cols
constexpr int kWG    = 4;        // ceil(127/32) w-groups of 32 pixels
constexpr int kWaves = 8;        // waves per 256-thread block (wave32)
}

__global__ __launch_bounds__(256) void eps_wmma_kernel(
    const float* __restrict__ in,    // (1, 128, 128, 128, 2) f32
    const float* __restrict__ core,  // (2,2,2,2,4) f32, 64 elems
    float* __restrict__ out) {       // (128, 127, 127, 4) f32
  // Core padded to 16x16 (cols 4..15 = 0) so B loads are unconditional.
  __shared__ float lds_core[16 * 16];
  __shared__ float lds_out[kWaves][32 * kOut];  // 32 pixels x 4 ch per wave

  const int tid = threadIdx.x;
  {
    const int row = tid >> 4;   // K index 0..15
    const int col = tid & 15;   // N index 0..15
    lds_core[tid] = (col < kOut) ? core[row * kOut + col] : 0.0f;
  }
  __syncthreads();

  const int wid  = tid >> 5;   // wave in block
  const int lane = tid & 31;
  const int m    = lane & 15;  // A-row (pixel-in-subgroup); B column (channel)
  const int half = lane >> 4;  // K-half selector per WMMA f32 16x16x4 layout

  const unsigned g   = (unsigned)blockIdx.x * kWaves + (unsigned)wid;
  const int      wg  = (int)(g & (kWG - 1));
  const unsigned rem = g >> 2;
  const int      h   = (int)(rem % (unsigned)kHo);
  const int      b   = (int)(rem / (unsigned)kHo);

  // Two pixel subgroups per wave: pixels wg*32 + m and wg*32 + 16 + m.
  const int w0 = wg * 32 + m;
  const int w1 = w0 + 16;
  const int wc0 = (w0 < kWo) ? w0 : (kWo - 1);
  const int wc1 = (w1 < kWo) ? w1 : (kWo - 1);

  const size_t rowbase = (size_t)(b * kH + h) * kW;

  // --- B chunks (4x16 f32, shared by both accumulators) ---
  // lane holds B[K = 4j + 2*half + v][N = m]
  const int kb = (2 * half) * 16 + m;
  v2f b0, b1, b2, b3;
  b0.x = lds_core[  0 + kb];  b0.y = lds_core[  0 + kb + 16];
  b1.x = lds_core[ 64 + kb];  b1.y = lds_core[ 64 + kb + 16];
  b2.x = lds_core[128 + kb];  b2.y = lds_core[128 + kb + 16];
  b3.x = lds_core[192 + kb];  b3.y = lds_core[192 + kb + 16];

  v8f acc0 = {0.f, 0.f, 0.f, 0.f, 0.f, 0.f, 0.f, 0.f};
  v8f acc1 = {0.f, 0.f, 0.f, 0.f, 0.f, 0.f, 0.f, 0.f};

#pragma unroll
  for (int sub = 0; sub < 2; ++sub) {
    const size_t base = (rowbase + (sub ? wc1 : wc0)) * kS;
    const v2f x00 = *(const v2f*)(in + base);
    const v2f x01 = *(const v2f*)(in + base + kS);
    const v2f x10 = *(const v2f*)(in + base + (size_t)kW * kS);
    const v2f x11 = *(const v2f*)(in + base + (size_t)kW * kS + kS);

    // w16[i] = x00[s0]x01[s1]x10[s2]x11[s3], i = s0*8+s1*4+s2*2+s3
    //        = p01[i>>2] * p23[i&3]; lane needs p23 only at {2*half, 2*half+1}
    float p01[4];
    p01[0] = x00.x * x01.x;
    p01[1] = x00.x * x01.y;
    p01[2] = x00.y * x01.x;
    p01[3] = x00.y * x01.y;
    const float s2v = half ? x10.y : x10.x;
    const float q0 = s2v * x11.x;
    const float q1 = s2v * x11.y;

    // A chunks (16x4 f32 layout): lane holds K = 4j + 2*half + {0,1}
    v2f a0, a1, a2, a3;
    a0.x = p01[0] * q0; a0.y = p01[0] * q1;
    a1.x = p01[1] * q0; a1.y = p01[1] * q1;
    a2.x = p01[2] * q0; a2.y = p01[2] * q1;
    a3.x = p01[3] * q0; a3.y = p01[3] * q1;

    v8f acc = sub ? acc1 : acc0;
    acc = __builtin_amdgcn_wmma_f32_16x16x4_f32(false, a0, false, b0, (short)0,
                                                acc, false, false);
    acc = __builtin_amdgcn_wmma_f32_16x16x4_f32(false, a1, false, b1, (short)0,
                                                acc, false, false);
    acc = __builtin_amdgcn_wmma_f32_16x16x4_f32(false, a2, false, b2, (short)0,
                                                acc, false, false);
    acc = __builtin_amdgcn_wmma_f32_16x16x4_f32(false, a3, false, b3, (short)0,
                                                acc, false, false);
    if (sub) acc1 = acc; else acc0 = acc;
  }

  // D layout: lane holds D[M = r + 8*half][N = m] in acc[r].
  // Transpose through LDS so every lane emits one coalesced 16B pixel store.
  if (m < kOut) {
#pragma unroll
    for (int r = 0; r < 8; ++r) {
      lds_out[wid][(r + 8 * half) * kOut + m]      = acc0[r];
      lds_out[wid][(16 + r + 8 * half) * kOut + m] = acc1[r];
    }
  }
  asm volatile("s_wait_dscnt 0" ::: "memory");  // wave-local LDS RAW fence

  // Lane l stores pixel wg*32 + l (all 32 lanes -> 512B contiguous per wave).
  const int wl = wg * 32 + lane;
  if (wl < kWo) {
    const v4f o = *(const v4f*)&lds_out[wid][lane * kOut];
    *(v4f*)(out + (((size_t)b * kHo + h) * kWo + wl) * kOut) = o;
  }
}

extern "C" void kernel_launch(void* const* d_in, const int* in_sizes, int n_in,
                              void* d_out, int out_size, void* d_ws, size_t ws_size,
                              hipStream_t stream) {
  (void)in_sizes; (void)n_in; (void)out_size; (void)d_ws; (void)ws_size;
  const float* in   = (const float*)d_in[0];
  const float* core = (const float*)d_in[1];
  float*       out  = (float*)d_out;

  const int total_waves = kB * kHo * kWG;       // 128*127*4 = 65024 (exact)
  const int blocks      = total_waves / kWaves; // 8128
  eps_wmma_kernel<<<blocks, 256, 0, stream>>>(in, core, out);
}